// BEANConvSample_13941463842867
// MI455X (gfx1250) — compile-verified
//
#include <hip/hip_runtime.h>
#include <hip/hip_bf16.h>

typedef __attribute__((ext_vector_type(2))) float v2f;
typedef __attribute__((ext_vector_type(8))) float v8f;

#define EPS_BN 1e-5f
#define MAXK 896   // largest concat-K of the three GEMMs

// ---- ordered-uint mapping for float atomic max (monotone) ----
__device__ __forceinline__ unsigned f2ou(float f) {
    unsigned u = __float_as_uint(f);
    return (u >> 31) ? ~u : (u | 0x80000000u);
}
__device__ __forceinline__ float ou2f(unsigned u) {
    return (u >> 31) ? __uint_as_float(u & 0x7fffffffu) : __uint_as_float(~u);
}
// f2ou(-inf) == 0x007FFFFF (host-side init constant)
#define OU_NEG_INF 0x007FFFFFu

__device__ __forceinline__ v8f wmma4(v2f a, v2f b, v8f c) {
    return __builtin_amdgcn_wmma_f32_16x16x4_f32(false, a, false, b, (short)0, c, false, false);
}

// ---------------------------------------------------------------------------
// fill
__global__ __launch_bounds__(256) void k_fill(unsigned* __restrict__ p, unsigned v, size_t n) {
    size_t i = (size_t)blockIdx.x * 256 + threadIdx.x;
    if (i < n) p[i] = v;
}

// ---------------------------------------------------------------------------
// logits: l[i] = dot(xs[i,:256], attn); track global max (ordered uint)
__global__ __launch_bounds__(256) void k_logits(const float* __restrict__ xs,
                                                const float* __restrict__ attn,
                                                float* __restrict__ logits,
                                                unsigned* __restrict__ gmax, int n) {
    int lane = threadIdx.x & 31, wave = threadIdx.x >> 5;
    int node = blockIdx.x * 8 + wave;
    if (node >= n) return;
    const float* row = xs + (size_t)node * 256;
    float s = 0.f;
    for (int f = lane; f < 256; f += 32) s += row[f] * attn[f];
    for (int m = 16; m; m >>= 1) s += __shfl_xor(s, m, 32);
    if (lane == 0) { logits[node] = s; atomicMax(gmax, f2ou(s)); }
}

// softmax denom: logits -> exp(l - gmax) in place; denom += exp
__global__ __launch_bounds__(256) void k_denom(float* __restrict__ logits,
                                               const unsigned* __restrict__ gmax,
                                               float* __restrict__ denom, int n) {
    int i = blockIdx.x * 256 + threadIdx.x;
    float e = 0.f;
    if (i < n) { e = __expf(logits[i] - ou2f(*gmax)); logits[i] = e; }
    for (int m = 16; m; m >>= 1) e += __shfl_xor(e, m, 32);
    if ((threadIdx.x & 31) == 0) atomicAdd(denom, e);
}

// xw[i,f] = (exp_i/denom) * xs[i,f]
__global__ __launch_bounds__(256) void k_weight(const float* __restrict__ xs,
                                                const float* __restrict__ exps,
                                                const float* __restrict__ denom,
                                                float* __restrict__ xw, size_t total) {
    size_t i = (size_t)blockIdx.x * 256 + threadIdx.x;
    if (i < total) { int node = (int)(i >> 8); xw[i] = xs[i] * (exps[node] / denom[0]); }
}

// ---------------------------------------------------------------------------
// per-edge scatter: one 256-thread block per edge
__global__ __launch_bounds__(256) void k_scatter(const float* __restrict__ xw,    // [nsrc,256]
                                                 const float* __restrict__ xe,    // [E,64]
                                                 const int* __restrict__ src_idx, // source node per edge
                                                 const int* __restrict__ dst_idx, // segment per edge
                                                 float* __restrict__ sum_n, unsigned* __restrict__ max_n,
                                                 float* __restrict__ sum_e, unsigned* __restrict__ max_e,
                                                 float* __restrict__ cnt) {
    int e = blockIdx.x, t = threadIdx.x;
    int s = src_idx[e], d = dst_idx[e];
    float v = xw[(size_t)s * 256 + t];
    atomicAdd(&sum_n[(size_t)d * 256 + t], v);
    atomicMax(&max_n[(size_t)d * 256 + t], f2ou(v));
    if (t < 64) {
        float w = xe[(size_t)e * 64 + t];
        atomicAdd(&sum_e[(size_t)d * 64 + t], w);
        atomicMax(&max_e[(size_t)d * 64 + t], f2ou(w));
    }
    if (t == 0) atomicAdd(&cnt[d], 1.0f);
}

// finalize: sum -> mean (div by max(cnt,1)); max ordered-uint -> float, empty seg -> 0
__global__ __launch_bounds__(256) void k_finalize(float* __restrict__ sum_n, unsigned* __restrict__ max_n,
                                                  float* __restrict__ sum_e, unsigned* __restrict__ max_e,
                                                  const float* __restrict__ cnt, int n) {
    size_t i = (size_t)blockIdx.x * 256 + threadIdx.x;
    if (i >= (size_t)n * 256) return;
    int node = (int)(i >> 8), f = (int)(i & 255);
    float c = cnt[node];
    float inv = 1.f / fmaxf(c, 1.f);
    sum_n[i] *= inv;
    ((float*)max_n)[i] = (c > 0.f) ? ou2f(max_n[i]) : 0.f;
    if (f < 64) {
        size_t j = (size_t)node * 64 + f;
        sum_e[j] *= inv;
        ((float*)max_e)[j] = (c > 0.f) ? ou2f(max_e[j]) : 0.f;
    }
}

// ---------------------------------------------------------------------------
// Generic fp32 WMMA GEMM: out[M,N] = concat(regions)[M,K] @ W[K,N] + bias
// Block = 256 threads = 8 waves; wave w computes the 16x16 tile at rows
// m0 = bx*128 + 16*w, cols n0 = by*16. K stepped by 4 via V_WMMA_F32_16X16X4_F32,
// unrolled x4 so 4 A-loads (global_load_b64) and 4 B-loads (ds_load_b64) are in
// flight per iteration.
// The 16-column W panel for the whole K is staged in LDS once per block with a
// swizzled layout sB[(k>>1)*32 + n*2 + (k&1)] so each lane's (k,k+1) pair is one
// 8-byte conflict-free ds_load_b64 (lanes 0-15 -> banks 0-31, 16-31 -> 32-63).
// Fragment layout (ISA 7.12.2, 32-bit): lane (l = lane&15, hi = lane>>4):
//   A vgpr{0,1} = A[m0+l][k+2*hi + {0,1}]   B vgpr{0,1} = W[k+2*hi + {0,1}][n0+l]
//   C vgpr r    = C[m0 + r + 8*hi][n0+l]
__global__ __launch_bounds__(256)
void k_gemm(const float* p0, const int* i0, int ld0, int kw0,
            const float* p1, const int* i1, int ld1, int kw1,
            const float* p2, const int* i2, int ld2, int kw2,
            const float* p3, const int* i3, int ld3, int kw3,
            const float* p4, const int* i4, int ld4, int kw4,
            const float* __restrict__ W, const float* __restrict__ bias,
            float* __restrict__ out, int M, int N) {
    __shared__ float sB[MAXK * 16];

    const int Ktot = kw0 + kw1 + kw2 + kw3 + kw4;
    const int n0 = blockIdx.y * 16;

    // cooperative, coalesced W-panel stage (swizzled for b64 LDS reads)
    for (int idx = threadIdx.x; idx < Ktot * 16; idx += 256) {
        int k = idx >> 4, n = idx & 15;
        sB[(k >> 1) * 32 + n * 2 + (k & 1)] = W[(size_t)k * N + n0 + n];
    }
    __syncthreads();

    const int lane = threadIdx.x & 31;
    const int wave = threadIdx.x >> 5;
    const int l = lane & 15, hi = lane >> 4;
    const int m0 = blockIdx.x * 128 + wave * 16;
    int mrow = m0 + l;
    if (mrow >= M) mrow = M - 1;   // clamp (stores are masked); keeps EXEC all-ones for WMMA

    v8f c = {0.f, 0.f, 0.f, 0.f, 0.f, 0.f, 0.f, 0.f};
    int kbase = 0;

#define GEMM_REGION(P, I, LD, KW)                                                   \
    if ((KW) > 0) {                                                                 \
        const float* ap = (P) + (size_t)((I) ? (I)[mrow] : mrow) * (LD) + 2 * hi;   \
        const float* bp = sB + (size_t)((kbase >> 1) + hi) * 32 + 2 * l;            \
        for (int k = 0; k < (KW); k += 16) {                                        \
            v2f a0 = *(const v2f*)(ap + k);                                         \
            v2f a1 = *(const v2f*)(ap + k + 4);                                     \
            v2f a2 = *(const v2f*)(ap + k + 8);                                     \
            v2f a3 = *(const v2f*)(ap + k + 12);                                    \
            const float* bk = bp + (k >> 1) * 32;                                   \
            v2f b0 = *(const v2f*)(bk);                                             \
            v2f b1 = *(const v2f*)(bk + 64);                                        \
            v2f b2 = *(const v2f*)(bk + 128);                                       \
            v2f b3 = *(const v2f*)(bk + 192);                                       \
            c = wmma4(a0, b0, c);                                                   \
            c = wmma4(a1, b1, c);                                                   \
            c = wmma4(a2, b2, c);                                                   \
            c = wmma4(a3, b3, c);                                                   \
        }                                                                           \
        kbase += (KW);                                                              \
    }

    GEMM_REGION(p0, i0, ld0, kw0)
    GEMM_REGION(p1, i1, ld1, kw1)
    GEMM_REGION(p2, i2, ld2, kw2)
    GEMM_REGION(p3, i3, ld3, kw3)
    GEMM_REGION(p4, i4, ld4, kw4)
#undef GEMM_REGION

    float bb = bias[n0 + l];
    for (int r = 0; r < 8; ++r) {
        int m = m0 + r + 8 * hi;
        if (m < M) out[(size_t)m * N + (n0 + l)] = c[r] + bb;
    }
}

// ---------------------------------------------------------------------------
// BN column stats: stats[0:C] += sum, stats[C:2C] += sumsq (block-local partials)
__global__ __launch_bounds__(256) void k_bn_stats(const float* __restrict__ x, float* __restrict__ stats,
                                                  int R, int C, int rowsPerBlock) {
    int t = threadIdx.x;
    int col = t % C;
    int r0 = blockIdx.x * rowsPerBlock;
    int rend = min(r0 + rowsPerBlock, R);
    float s = 0.f, ss = 0.f;
    for (int r = r0 + t / C; r < rend; r += 256 / C) {
        float v = x[(size_t)r * C + col];
        s += v; ss += v * v;
    }
    atomicAdd(&stats[col], s);
    atomicAdd(&stats[C + col], ss);
}

// BN apply in place: (x - mu) * rsqrt(var + eps) * g + b
__global__ __launch_bounds__(256) void k_bn_apply(float* __restrict__ x, const float* __restrict__ stats,
                                                  const float* __restrict__ g, const float* __restrict__ b,
                                                  int R, int C) {
    size_t i = (size_t)blockIdx.x * 256 + threadIdx.x;
    if (i >= (size_t)R * C) return;
    int col = (int)(i % C);
    float invR = 1.f / (float)R;
    float mu = stats[col] * invR;
    float var = stats[C + col] * invR - mu * mu;
    float sc = rsqrtf(var + EPS_BN) * g[col];
    x[i] = (x[i] - mu) * sc + b[col];
}

// ---------------------------------------------------------------------------
extern "C" void kernel_launch(void* const* d_in, const int* in_sizes, int n_in,
                              void* d_out, int out_size, void* d_ws, size_t ws_size,
                              hipStream_t stream) {
    const float* xus = (const float*)d_in[0];
    const float* xut = (const float*)d_in[1];
    const float* xvs = (const float*)d_in[2];
    const float* xvt = (const float*)d_in[3];
    const float* xe_e   = (const float*)d_in[4];
    const float* xe_v2u = (const float*)d_in[5];
    const float* xe_u2v = (const float*)d_in[6];
    const int* row_v2u = (const int*)d_in[7];
    const int* col_v2u = (const int*)d_in[8];
    const int* row_u2v = (const int*)d_in[9];
    const int* col_u2v = (const int*)d_in[10];
    const int* row_e   = (const int*)d_in[11];
    const int* col_e   = (const int*)d_in[12];
    const float* attn_v2u = (const float*)d_in[13];
    const float* W_v2u = (const float*)d_in[14];
    const float* b_v2u = (const float*)d_in[15];
    const float* g_v2u = (const float*)d_in[16];
    const float* be_v2u = (const float*)d_in[17];
    const float* attn_u2v = (const float*)d_in[18];
    const float* W_u2v = (const float*)d_in[19];
    const float* b_u2v = (const float*)d_in[20];
    const float* g_u2v = (const float*)d_in[21];
    const float* be_u2v = (const float*)d_in[22];
    const float* W_e  = (const float*)d_in[23];
    const float* b_e  = (const float*)d_in[24];
    const float* g_e  = (const float*)d_in[25];
    const float* be_e = (const float*)d_in[26];

    const int NU = in_sizes[0] / 256;
    const int NV = in_sizes[2] / 256;
    const int E  = in_sizes[4] / 64;

    // ---- workspace layout (floats) ----
    float* ws = (float*)d_ws;
    size_t o = 0;
    auto alloc = [&](size_t n) { float* p = ws + o; o += n; return p; };
    float* xw_v = alloc((size_t)NV * 256);
    float* xw_u = alloc((size_t)NU * 256);
    // zero-init block
    float* zero0 = ws + o;
    float* sum_v  = alloc((size_t)NU * 256);
    float* sum_ev = alloc((size_t)NU * 64);
    float* cnt_u  = alloc(NU);
    float* sum_u  = alloc((size_t)NV * 256);
    float* sum_eu = alloc((size_t)NV * 64);
    float* cnt_v  = alloc(NV);
    float* logit_v = alloc(NV);
    float* logit_u = alloc(NU);
    float* denom_v = alloc(1);
    float* denom_u = alloc(1);
    float* bn_u = alloc(512);
    float* bn_v = alloc(512);
    float* bn_e = alloc(128);
    size_t nzero = (size_t)(ws + o - zero0);
    // -inf-init block (ordered-uint encoded)
    unsigned* ninf0 = (unsigned*)(ws + o);
    unsigned* max_v  = (unsigned*)alloc((size_t)NU * 256);
    unsigned* max_ev = (unsigned*)alloc((size_t)NU * 64);
    unsigned* max_u  = (unsigned*)alloc((size_t)NV * 256);
    unsigned* max_eu = (unsigned*)alloc((size_t)NV * 64);
    unsigned* gmax_v = (unsigned*)alloc(1);
    unsigned* gmax_u = (unsigned*)alloc(1);
    size_t nninf = (size_t)((unsigned*)(ws + o) - ninf0);
    (void)ws_size;

    auto nb = [](size_t n, int b) { return (unsigned)((n + (size_t)b - 1) / (size_t)b); };

    // ---- init accumulators ----
    k_fill<<<nb(nzero, 256), 256, 0, stream>>>((unsigned*)zero0, 0u, nzero);
    k_fill<<<nb(nninf, 256), 256, 0, stream>>>(ninf0, OU_NEG_INF, nninf);

    // ---- attention softmax weights ----
    k_logits<<<nb(NV, 8), 256, 0, stream>>>(xvs, attn_v2u, logit_v, gmax_v, NV);
    k_logits<<<nb(NU, 8), 256, 0, stream>>>(xus, attn_u2v, logit_u, gmax_u, NU);
    k_denom<<<nb(NV, 256), 256, 0, stream>>>(logit_v, gmax_v, denom_v, NV);
    k_denom<<<nb(NU, 256), 256, 0, stream>>>(logit_u, gmax_u, denom_u, NU);
    k_weight<<<nb((size_t)NV * 256, 256), 256, 0, stream>>>(xvs, logit_v, denom_v, xw_v, (size_t)NV * 256);
    k_weight<<<nb((size_t)NU * 256, 256), 256, 0, stream>>>(xus, logit_u, denom_u, xw_u, (size_t)NU * 256);

    // ---- segment scatter (mean/max numerators) ----
    // v->u: msg = xw_v[col_v2u], segment by row_v2u
    k_scatter<<<E, 256, 0, stream>>>(xw_v, xe_v2u, col_v2u, row_v2u, sum_v, max_v, sum_ev, max_ev, cnt_u);
    // u->v: msg = xw_u[row_u2v], segment by col_u2v
    k_scatter<<<E, 256, 0, stream>>>(xw_u, xe_u2v, row_u2v, col_u2v, sum_u, max_u, sum_eu, max_eu, cnt_v);
    k_finalize<<<nb((size_t)NU * 256, 256), 256, 0, stream>>>(sum_v, max_v, sum_ev, max_ev, cnt_u, NU);
    k_finalize<<<nb((size_t)NV * 256, 256), 256, 0, stream>>>(sum_u, max_u, sum_eu, max_eu, cnt_v, NV);

    // ---- output partitions ----
    float* out_u = (float*)d_out;
    float* out_v = out_u + (size_t)NU * 256;
    float* out_e = out_v + (size_t)NV * 256;

    // ---- WMMA GEMMs ----
    // out_u = [xut | mean_v | max_v | mean_ev | max_ev] @ W_v2u + b
    k_gemm<<<dim3(nb(NU, 128), 16), 256, 0, stream>>>(
        xut, nullptr, 256, 256,
        sum_v, nullptr, 256, 256,
        (const float*)max_v, nullptr, 256, 256,
        sum_ev, nullptr, 64, 64,
        (const float*)max_ev, nullptr, 64, 64,
        W_v2u, b_v2u, out_u, NU, 256);
    // out_v = [xvt | mean_u | max_u | mean_eu | max_eu] @ W_u2v + b
    k_gemm<<<dim3(nb(NV, 128), 16), 256, 0, stream>>>(
        xvt, nullptr, 256, 256,
        sum_u, nullptr, 256, 256,
        (const float*)max_u, nullptr, 256, 256,
        sum_eu, nullptr, 64, 64,
        (const float*)max_eu, nullptr, 64, 64,
        W_u2v, b_u2v, out_v, NV, 256);
    // out_e = [xe_e | xut[row_e] | xvt[col_e]] @ W_e + b (gathered A rows)
    k_gemm<<<dim3(nb(E, 128), 4), 256, 0, stream>>>(
        xe_e, nullptr, 64, 64,
        xut, row_e, 256, 256,
        xvt, col_e, 256, 256,
        nullptr, nullptr, 0, 0,
        nullptr, nullptr, 0, 0,
        W_e, b_e, out_e, E, 64);

    // ---- batchnorm ----
    k_bn_stats<<<nb(NU, 128), 256, 0, stream>>>(out_u, bn_u, NU, 256, 128);
    k_bn_stats<<<nb(NV, 128), 256, 0, stream>>>(out_v, bn_v, NV, 256, 128);
    k_bn_stats<<<nb(E, 512), 256, 0, stream>>>(out_e, bn_e, E, 64, 512);
    k_bn_apply<<<nb((size_t)NU * 256, 256), 256, 0, stream>>>(out_u, bn_u, g_v2u, be_v2u, NU, 256);
    k_bn_apply<<<nb((size_t)NV * 256, 256), 256, 0, stream>>>(out_v, bn_v, g_u2v, be_u2v, NV, 256);
    k_bn_apply<<<nb((size_t)E * 64, 256), 256, 0, stream>>>(out_e, bn_e, g_e, be_e, E, 64);
}